// SimplePMA_15522011807840
// MI455X (gfx1250) — compile-verified
//
#include <hip/hip_runtime.h>
#include <hip/hip_bf16.h>

// PMA with single learned query, algebraically collapsed:
//   scores = SCALE * (x @ P + sbias)   with P = Wk_headslices @ q  (512x8)
//   y[b]   = attn[b] @ x[b]            (8x4096 @ 4096x512, fp32 WMMA, K split 8 ways)
//   out    = (y @ Wv_slices + bv) @ Wo + bo
// Memory-bound: 2 passes over x (1 GB) ~= 43 us at 23.3 TB/s.

#define DIM      512
#define HEADS    8
#define HEAD_DIM 64
#define BB       64
#define LL       4096
#define SCALE    0.125f
#define KSPLIT   8
#define KCH      (LL / KSPLIT)   // 512 K-rows per pooling block
#define ASTRIDE  17              // LDS row stride (gcd(17,64)=1 -> conflict-free)

typedef float v2f __attribute__((ext_vector_type(2)));
typedef float v8f __attribute__((ext_vector_type(8)));

// Workspace layout (floats):
#define WS_P   0                        // P[h*512+d], 4096 floats
#define WS_SB  4096                     // sbias[8]
#define WS_SC  8192                     // scores/attn [B][H][L] = 2,097,152
#define WS_Y   (8192 + BB*HEADS*LL)     // y partials [B][KSPLIT][H][DIM] = 2,097,152

// ---------------- Kernel A: q = seed@Wq+bq ; P = Wk_slice@q ; sbias = q.bk ----
__global__ void __launch_bounds__(512) pma_setup(
    const float* __restrict__ seed, const float* __restrict__ Wq,
    const float* __restrict__ bq,   const float* __restrict__ Wk,
    const float* __restrict__ bk,   float* __restrict__ ws) {
  __shared__ float qs[DIM];
  const int t = threadIdx.x;  // 512
  float acc = bq[t];
  for (int d = 0; d < DIM; ++d) acc += seed[d] * Wq[d * DIM + t];
  qs[t] = acc;
  __syncthreads();
  for (int h = 0; h < HEADS; ++h) {
    float p = 0.f;
    const float* wrow = Wk + t * DIM + h * HEAD_DIM;
    const float* qh = qs + h * HEAD_DIM;
    #pragma unroll 8
    for (int c = 0; c < HEAD_DIM; ++c) p += wrow[c] * qh[c];
    ws[WS_P + h * DIM + t] = p;
  }
  if (t < HEADS) {
    float sb = 0.f;
    for (int c = 0; c < HEAD_DIM; ++c) sb += qs[t * HEAD_DIM + c] * bk[t * HEAD_DIM + c];
    ws[WS_SB + t] = sb;
  }
}

// ---------------- Kernel B: scores[b,h,l] = SCALE*(x[b,l].P[:,h] + sbias[h]) --
__global__ void __launch_bounds__(512) pma_scores(
    const float* __restrict__ x, const unsigned char* __restrict__ mask,
    const float* __restrict__ ws, float* __restrict__ scores) {
  __shared__ float pLds[HEADS * DIM];
  __shared__ float sbLds[HEADS];
  const int t = threadIdx.x;       // 512 = 16 waves
  const int b = blockIdx.y;
  for (int i = t; i < HEADS * DIM; i += 512) pLds[i] = ws[WS_P + i];
  if (t < HEADS) sbLds[t] = ws[WS_SB + t];
  __syncthreads();
  const int wave = t >> 5, lane = t & 31;
  const int l0 = blockIdx.x * 64 + wave * 4;
  for (int r = 0; r < 4; ++r) {
    const int l = l0 + r;
    const float* xr = x + ((size_t)b * LL + l) * DIM;
    float p[HEADS] = {0.f, 0.f, 0.f, 0.f, 0.f, 0.f, 0.f, 0.f};
    #pragma unroll 4
    for (int tt = 0; tt < 16; ++tt) {
      const int idx = lane + 32 * tt;           // coalesced, LDS conflict-free
      const float xv = xr[idx];
      #pragma unroll
      for (int h = 0; h < HEADS; ++h) p[h] += xv * pLds[h * DIM + idx];
    }
    #pragma unroll
    for (int off = 16; off >= 1; off >>= 1)
      #pragma unroll
      for (int h = 0; h < HEADS; ++h) p[h] += __shfl_xor(p[h], off, 32);
    if (lane == 0) {
      const bool mv = mask[(size_t)b * LL + l] != 0;
      #pragma unroll
      for (int h = 0; h < HEADS; ++h) {
        float s = SCALE * (p[h] + sbLds[h]);
        if (!mv) s = -__builtin_inff();
        scores[((size_t)(b * HEADS + h)) * LL + l] = s;
      }
    }
  }
}

// ---------------- Kernel C: softmax over L per (b,h), in place ----------------
__global__ void __launch_bounds__(256) pma_softmax(float* __restrict__ attn) {
  float* row = attn + (size_t)blockIdx.x * LL;  // 512 blocks = B*H
  const int t = threadIdx.x;                    // 256
  __shared__ float red[256];
  float v[16];
  float m = -__builtin_inff();
  #pragma unroll
  for (int i = 0; i < 16; ++i) { v[i] = row[t + 256 * i]; m = fmaxf(m, v[i]); }
  red[t] = m; __syncthreads();
  for (int s = 128; s >= 1; s >>= 1) { if (t < s) red[t] = fmaxf(red[t], red[t + s]); __syncthreads(); }
  m = red[0]; __syncthreads();
  float sum = 0.f;
  #pragma unroll
  for (int i = 0; i < 16; ++i) { v[i] = __expf(v[i] - m); sum += v[i]; }
  red[t] = sum; __syncthreads();
  for (int s = 128; s >= 1; s >>= 1) { if (t < s) red[t] += red[t + s]; __syncthreads(); }
  const float inv = 1.0f / red[0];
  #pragma unroll
  for (int i = 0; i < 16; ++i) row[t + 256 * i] = v[i] * inv;
}

// ---------------- Kernel D: y_part[b,ks] = attn_pad[16xKCH] @ x[b][KCH x 512] -
// fp32 WMMA 16x16x4. A rows 0..7 = heads, rows 8..15 = zeros (padded in LDS, so
// fragment loads are UNCONDITIONAL ds_load_b32 -> no exec-mask juggling).
// A layout: lane m=lane&15 holds row m; K split {k0,k0+1}/{k0+2,k0+3} by lane half.
// B layout: VGPR0 = rows {k0, k0+2}, VGPR1 = rows {k0+1, k0+3} striped over lanes.
__global__ void __launch_bounds__(1024) pma_pool_wmma(
    const float* __restrict__ x, const float* __restrict__ attn,
    float* __restrict__ y) {
  extern __shared__ float aT[];                 // [KCH][ASTRIDE] = 34,816 B
  const int t  = threadIdx.x;                   // 1024 = 32 waves
  const int b  = blockIdx.x;                    // 64
  const int ks = blockIdx.y;                    // 8
  const int kbase = ks * KCH;
  const float* arow = attn + (size_t)b * HEADS * LL + kbase;
  // zero-fill (covers pad rows 8..15), then transpose-stage the attn chunk
  for (int i = t; i < KCH * ASTRIDE; i += 1024) aT[i] = 0.f;
  __syncthreads();
  for (int i = t; i < HEADS * KCH; i += 1024) {
    const int h = i >> 9, l = i & (KCH - 1);
    aT[l * ASTRIDE + h] = arow[(size_t)h * LL + l];
  }
  __syncthreads();
  const int wave = t >> 5, lane = t & 31;
  const int m = lane & 15, hi = lane >> 4;
  const int n = wave * 16 + m;                  // this wave's N-tile column
  const float* xb = x + ((size_t)b * LL + kbase) * DIM;
  v8f c = {0.f, 0.f, 0.f, 0.f, 0.f, 0.f, 0.f, 0.f};
  for (int k0 = 0; k0 < KCH; k0 += 16) {        // 4 WMMAs per trip, loads batched
    v2f A[4], Bv[4];
    #pragma unroll
    for (int u = 0; u < 4; ++u) {
      const int kk = k0 + 4 * u + 2 * hi;
      A[u].x  = aT[kk * ASTRIDE + m];           // unconditional, conflict-free
      A[u].y  = aT[(kk + 1) * ASTRIDE + m];
      Bv[u].x = xb[(size_t)kk * DIM + n];       // 64B/segment coalesced
      Bv[u].y = xb[(size_t)(kk + 1) * DIM + n];
    }
    #pragma unroll
    for (int u = 0; u < 4; ++u)
      c = __builtin_amdgcn_wmma_f32_16x16x4_f32(
          false, A[u], false, Bv[u], (short)0, c, false, false);
  }
  if (lane < 16) {                              // rows 0..7 in c[0..7], lanes 0..15
    #pragma unroll
    for (int v = 0; v < HEADS; ++v)
      y[(((size_t)b * KSPLIT + ks) * HEADS + v) * DIM + n] = c[v];
  }
}

// ------- Kernel E: reduce y partials; out = (y @ Wv_slices + bv) @ Wo + bo ----
__global__ void __launch_bounds__(512) pma_out(
    const float* __restrict__ y,  const float* __restrict__ Wv,
    const float* __restrict__ bv, const float* __restrict__ Wo,
    const float* __restrict__ bo, float* __restrict__ out) {
  __shared__ float ys[HEADS * DIM];   // reduced y[b]  (16 KB)
  __shared__ float o1[DIM];
  const int t = threadIdx.x;  // 512
  const int b = blockIdx.x;   // 64
  for (int i = t; i < HEADS * DIM; i += 512) {
    const int h = i >> 9, d = i & (DIM - 1);
    float s = 0.f;
    #pragma unroll
    for (int ks = 0; ks < KSPLIT; ++ks)
      s += y[(((size_t)b * KSPLIT + ks) * HEADS + h) * DIM + d];
    ys[i] = s;
  }
  __syncthreads();
  const int h = t >> 6;       // head owning output column t
  float acc = bv[t];
  for (int d = 0; d < DIM; ++d) acc += ys[h * DIM + d] * Wv[d * DIM + t];
  o1[t] = acc;
  __syncthreads();
  float o = bo[t];
  for (int cidx = 0; cidx < DIM; ++cidx) o += o1[cidx] * Wo[cidx * DIM + t];
  out[(size_t)b * DIM + t] = o;
}

extern "C" void kernel_launch(void* const* d_in, const int* in_sizes, int n_in,
                              void* d_out, int out_size, void* d_ws, size_t ws_size,
                              hipStream_t stream) {
  const float* x          = (const float*)d_in[0];
  const unsigned char* mk = (const unsigned char*)d_in[1];  // bool mask
  const float* seed       = (const float*)d_in[2];
  const float* Wq         = (const float*)d_in[3];
  const float* bq         = (const float*)d_in[4];
  const float* Wk         = (const float*)d_in[5];
  const float* bk         = (const float*)d_in[6];
  const float* Wv         = (const float*)d_in[7];
  const float* bv         = (const float*)d_in[8];
  const float* Wo         = (const float*)d_in[9];
  const float* bo         = (const float*)d_in[10];
  float* ws      = (float*)d_ws;
  float* scores  = ws + WS_SC;
  float* y       = ws + WS_Y;
  float* out     = (float*)d_out;

  pma_setup   <<<1, 512, 0, stream>>>(seed, Wq, bq, Wk, bk, ws);
  pma_scores  <<<dim3(LL / 64, BB), 512, 0, stream>>>(x, mk, ws, scores);
  pma_softmax <<<BB * HEADS, 256, 0, stream>>>(scores);
  pma_pool_wmma<<<dim3(BB, KSPLIT), 1024, KCH * ASTRIDE * sizeof(float), stream>>>(x, scores, y);
  pma_out     <<<BB, 512, 0, stream>>>(y, Wv, bv, Wo, bo, out);
}